// HeteroLightGCN_56556129354073
// MI455X (gfx1250) — compile-verified
//
#include <hip/hip_runtime.h>

// HeteroLightGCN on MI455X (gfx1250).
//
// Pure gather/scatter-add workload (no GEMM content -> WMMA inapplicable).
// Strategy: fold degree-normalization + relation weights into per-source
// scale arrays computed once; scatter with native fp32 L2 atomics
// (global_atomic_add_f32 scope:SCOPE_DEV). All mutable aggregation buffers
// (~115 MB) fit in the 192 MB L2, so atomics RMW at cache bandwidth.
// Edge index streams are marked non-temporal so they don't evict the
// L2-resident embedding tables.

#define EMB 64
static constexpr int TPB = 256;

// Guaranteed-native gfx1250 fp32 global atomic add, no return (STOREcnt),
// device scope so all WGPs RMW coherently at L2.
__device__ __forceinline__ void atom_add_f32(float* p, float v) {
  asm volatile("global_atomic_add_f32 %0, %1, off scope:SCOPE_DEV"
               :: "v"((unsigned long long)p), "v"(v)
               : "memory");
}

__global__ void k_fill0(float4* __restrict__ p, int n4) {
  int i = blockIdx.x * blockDim.x + threadIdx.x;
  if (i < n4) p[i] = make_float4(0.f, 0.f, 0.f, 0.f);
}

__global__ void k_copy(float4* __restrict__ d, const float4* __restrict__ s, int n4) {
  int i = blockIdx.x * blockDim.x + threadIdx.x;
  if (i < n4) d[i] = s[i];
}

// out = (out + x) * scale    (scale=1 for running sum, 0.25 fused on last layer)
__global__ void k_accum(float4* __restrict__ out, const float4* __restrict__ x,
                        int n4, float scale) {
  int i = blockIdx.x * blockDim.x + threadIdx.x;
  if (i < n4) {
    float4 o = out[i];
    float4 v = x[i];
    o.x = (o.x + v.x) * scale;
    o.y = (o.y + v.y) * scale;
    o.z = (o.z + v.z) * scale;
    o.w = (o.w + v.w) * scale;
    out[i] = o;
  }
}

// Degree count: cnt[src[e]] += 1.0f   (index stream is non-temporal)
__global__ void k_count(const int* __restrict__ src, int nE, float* __restrict__ cnt) {
  int e = blockIdx.x * blockDim.x + threadIdx.x;
  if (e < nE) atom_add_f32(cnt + __builtin_nontemporal_load(src + e), 1.0f);
}

// cnt[i] = weight / max(cnt[i], 1)   (folds relation weight into inv-degree)
__global__ void k_invert(float* __restrict__ cnt, int n, float scale) {
  int i = blockIdx.x * blockDim.x + threadIdx.x;
  if (i < n) cnt[i] = scale / fmaxf(cnt[i], 1.0f);
}

// Edge scatter: 16 lanes per edge, each lane owns a float4 chunk of the
// 64-float row. b128 gather from tgt[dst] (RT policy: rows re-used, L2
// resident), 4 fp32 L2 atomics into agg[src], pre-scaled by
// inv[src] (= weight / degree). Edge index loads are non-temporal.
__global__ void k_edge(const int* __restrict__ src, const int* __restrict__ dst,
                       int nE, const float* __restrict__ inv,
                       const float* __restrict__ tgt, float* __restrict__ agg) {
  int t = blockIdx.x * blockDim.x + threadIdx.x;
  int e = t >> 4;
  if (e >= nE) return;
  int c = t & 15;
  int s = __builtin_nontemporal_load(src + e);
  int d = __builtin_nontemporal_load(dst + e);
  float w = inv[s];
  float4 v = *(const float4*)(tgt + (size_t)d * EMB + (size_t)c * 4);
  float* a = agg + (size_t)s * EMB + (size_t)c * 4;
  atom_add_f32(a + 0, v.x * w);
  atom_add_f32(a + 1, v.y * w);
  atom_add_f32(a + 2, v.z * w);
  atom_add_f32(a + 3, v.w * w);
}

extern "C" void kernel_launch(void* const* d_in, const int* in_sizes, int n_in,
                              void* d_out, int out_size, void* d_ws, size_t ws_size,
                              hipStream_t stream) {
  (void)n_in; (void)out_size; (void)ws_size;

  const float* cust_w  = (const float*)d_in[0];
  const float* prod_w  = (const float*)d_in[1];
  const float* group_w = (const float*)d_in[2];
  const float* brand_w = (const float*)d_in[3];
  const int* purch_src = (const int*)d_in[4];
  const int* purch_dst = (const int*)d_in[5];
  const int* pby_src   = (const int*)d_in[6];
  const int* pby_dst   = (const int*)d_in[7];
  const int* sim_src   = (const int*)d_in[8];
  const int* sim_dst   = (const int*)d_in[9];
  const int* cop_src   = (const int*)d_in[10];
  const int* cop_dst   = (const int*)d_in[11];
  const int* bel_src   = (const int*)d_in[12];
  const int* bel_dst   = (const int*)d_in[13];
  const int* comp_src  = (const int*)d_in[14];
  const int* comp_dst  = (const int*)d_in[15];

  const int NC = in_sizes[0] / EMB;   // 300000
  const int NP = in_sizes[1] / EMB;   // 150000
  const int NG = in_sizes[2] / EMB;   // 2000
  const int NB = in_sizes[3] / EMB;   // 500
  const int E_purch = in_sizes[4];
  const int E_pby   = in_sizes[6];
  const int E_sim   = in_sizes[8];
  const int E_cop   = in_sizes[10];
  const int E_bel   = in_sizes[12];
  const int E_comp  = in_sizes[14];

  // Workspace layout (floats): ping-pong embeddings + inverse-degree arrays.
  float* ws = (float*)d_ws;
  float* custA  = ws;  ws += (size_t)NC * EMB;
  float* custB  = ws;  ws += (size_t)NC * EMB;
  float* prodA  = ws;  ws += (size_t)NP * EMB;
  float* prodB  = ws;  ws += (size_t)NP * EMB;
  float* invC   = ws;  ws += NC;       // 1/deg  (purchased)
  float* invPby = ws;  ws += NP;       // 1/deg
  float* invSim = ws;  ws += NP;       // 0.5/deg
  float* invCop = ws;  ws += NP;       // 0.3/deg
  float* invBel = ws;  ws += NP;       // 0.2/deg
  float* invComp= ws;  ws += NP;       // 0.2/deg

  float* outC = (float*)d_out;
  float* outP = outC + (size_t)NC * EMB;
  float* outG = outP + (size_t)NP * EMB;
  float* outB = outG + (size_t)NG * EMB;

  auto g1 = [](long long n) { return (unsigned)((n + TPB - 1) / TPB); };
  auto ge = [](long long e) { return (unsigned)((e * 16 + TPB - 1) / TPB); };

  const int nC4 = NC * EMB / 4;
  const int nP4 = NP * EMB / 4;

  // ---- Degrees (static across layers): zero, count, invert+fold weight ----
  const int invTot = NC + 5 * NP;              // divisible by 4 for these sizes
  k_fill0<<<g1(invTot / 4), TPB, 0, stream>>>((float4*)invC, invTot / 4);
  k_count<<<g1(E_purch), TPB, 0, stream>>>(purch_src, E_purch, invC);
  k_count<<<g1(E_pby),   TPB, 0, stream>>>(pby_src,   E_pby,   invPby);
  k_count<<<g1(E_sim),   TPB, 0, stream>>>(sim_src,   E_sim,   invSim);
  k_count<<<g1(E_cop),   TPB, 0, stream>>>(cop_src,   E_cop,   invCop);
  k_count<<<g1(E_bel),   TPB, 0, stream>>>(bel_src,   E_bel,   invBel);
  k_count<<<g1(E_comp),  TPB, 0, stream>>>(comp_src,  E_comp,  invComp);
  k_invert<<<g1(NC), TPB, 0, stream>>>(invC,   NC, 1.0f);
  k_invert<<<g1(NP), TPB, 0, stream>>>(invPby, NP, 1.0f);
  k_invert<<<g1(NP), TPB, 0, stream>>>(invSim, NP, 0.5f);
  k_invert<<<g1(NP), TPB, 0, stream>>>(invCop, NP, 0.3f);
  k_invert<<<g1(NP), TPB, 0, stream>>>(invBel, NP, 0.2f);
  k_invert<<<g1(NP), TPB, 0, stream>>>(invComp,NP, 0.2f);

  // ---- Running layer-mean accumulators start at the layer-0 embeddings ----
  k_copy<<<g1(nC4), TPB, 0, stream>>>((float4*)outC, (const float4*)cust_w, nC4);
  k_copy<<<g1(nP4), TPB, 0, stream>>>((float4*)outP, (const float4*)prod_w, nP4);
  // group/brand never change: mean of identical copies == copy
  k_copy<<<g1(NG * EMB / 4), TPB, 0, stream>>>((float4*)outG, (const float4*)group_w, NG * EMB / 4);
  k_copy<<<g1(NB * EMB / 4), TPB, 0, stream>>>((float4*)outB, (const float4*)brand_w, NB * EMB / 4);

  // ---- 3 propagation layers (ping-pong, old embeddings feed the updates) ----
  const float* curC = cust_w;
  const float* curP = prod_w;
  for (int layer = 0; layer < 3; ++layer) {
    float* newC = (layer & 1) ? custB : custA;
    float* newP = (layer & 1) ? prodB : prodA;

    // newC = 0 (pure aggregation); newP = product_emb (the identity term)
    k_fill0<<<g1(nC4), TPB, 0, stream>>>((float4*)newC, nC4);
    k_copy <<<g1(nP4), TPB, 0, stream>>>((float4*)newP, (const float4*)curP, nP4);

    // customer <- mean over purchased products (old product emb)
    k_edge<<<ge(E_purch), TPB, 0, stream>>>(purch_src, purch_dst, E_purch, invC,    curP,    newC);
    // product += mean(old customer emb) + 0.5*sim + 0.3*cop + 0.2*group + 0.2*brand
    k_edge<<<ge(E_pby),   TPB, 0, stream>>>(pby_src,   pby_dst,   E_pby,   invPby,  curC,    newP);
    k_edge<<<ge(E_sim),   TPB, 0, stream>>>(sim_src,   sim_dst,   E_sim,   invSim,  curP,    newP);
    k_edge<<<ge(E_cop),   TPB, 0, stream>>>(cop_src,   cop_dst,   E_cop,   invCop,  curP,    newP);
    k_edge<<<ge(E_bel),   TPB, 0, stream>>>(bel_src,   bel_dst,   E_bel,   invBel,  group_w, newP);
    k_edge<<<ge(E_comp),  TPB, 0, stream>>>(comp_src,  comp_dst,  E_comp,  invComp, brand_w, newP);

    // accumulate into the mean; last layer fuses the /4
    const float s = (layer == 2) ? 0.25f : 1.0f;
    k_accum<<<g1(nC4), TPB, 0, stream>>>((float4*)outC, (const float4*)newC, nC4, s);
    k_accum<<<g1(nP4), TPB, 0, stream>>>((float4*)outP, (const float4*)newP, nP4, s);

    curC = newC;
    curP = newP;
  }
}